// CBOW_2594160247204
// MI455X (gfx1250) — compile-verified
//
#include <hip/hip_runtime.h>

// CBOW negative-sampling loss on gfx1250 (MI455X).
// Bandwidth-bound: ~176MB of gathered embedding rows -> ~7.5us floor @ 23.3 TB/s.
// Little's law (~700ns HBM latency) needs ~16MB in flight, so: 1 tile per block,
// 4 waves/block (4096 waves total), fully unrolled staging loops for deep
// global_load_b128 pipelines. Dots run on V_WMMA_F32_16X16X4_F32 (exact f32
// reference math); we consume the diagonal of each 16x16 WMMA tile.

typedef __attribute__((ext_vector_type(2))) float v2f;
typedef __attribute__((ext_vector_type(8))) float v8f;

#define DIMD 128
#define CTXW 10
#define NEGK 10
#define PAD  132              // LDS row pad (floats) -> conflict-free ds_load_2addr_b64
#define WAVES_PER_BLOCK 4
#define TPB (WAVES_PER_BLOCK * 32)

__device__ __forceinline__ float log_sigmoid(float x) {
    // stable: min(x,0) - log1p(exp(-|x|))
    return fminf(x, 0.0f) - log1pf(expf(-fabsf(x)));
}

__global__ void __launch_bounds__(TPB)
cbow_main(const int* __restrict__ context,   // [B, W]
          const int* __restrict__ target,    // [B]
          const int* __restrict__ noise,     // [B, K]
          const float* __restrict__ emb_w,   // [V, D]
          const float* __restrict__ ctx_w,   // [V, D]
          float* __restrict__ partial,       // [B/16] per-tile log-sigmoid sums
          int B) {
    __shared__ float cbuf[16 * PAD];                    // context means (block-shared)
    __shared__ float xbuf[WAVES_PER_BLOCK][16 * PAD];   // per-wave X rows
    __shared__ float wsum[WAVES_PER_BLOCK];

    const int wave = threadIdx.x >> 5;
    const int lane = threadIdx.x & 31;
    const int tile = blockIdx.x;            // one 16-row batch tile per block
    const int b0   = tile * 16;

    float* xb = xbuf[wave];
    const int h  = lane >> 4;   // lane half: selects K sub-pair {2h, 2h+1}
    const int mm = lane & 15;   // matrix row/col index this lane feeds

    // ---- Phase 1: context means, 4 rows per wave, fully unrolled (10 loads
    //      in flight per row). One 512B row per wave-wide float4 load. ----
    #pragma unroll
    for (int mi = 0; mi < 16 / WAVES_PER_BLOCK; ++mi) {
        const int m = wave * (16 / WAVES_PER_BLOCK) + mi;
        const int* cip = context + (size_t)(b0 + m) * CTXW;
        float4 a4 = make_float4(0.f, 0.f, 0.f, 0.f);
        #pragma unroll
        for (int w = 0; w < CTXW; ++w) {
            const int idx = cip[w];
            const float4 v = ((const float4*)(ctx_w + (size_t)idx * DIMD))[lane];
            a4.x += v.x; a4.y += v.y; a4.z += v.z; a4.w += v.w;
        }
        const float s = 1.0f / (float)CTXW;
        float* dst = cbuf + m * PAD + lane * 4;
        dst[0] = a4.x * s; dst[1] = a4.y * s; dst[2] = a4.z * s; dst[3] = a4.w * s;
    }
    __syncthreads();   // cbuf visible to all waves

    // ---- Phase 2: 11 score passes split round-robin across the 4 waves.
    //      (s==0: target, s>=1: noise k=s-1). Non-uniform trip count across
    //      waves -> wave-local LDS ordering only (DS ops are in-order per wave),
    //      each wave owns its xb buffer. ----
    float lsum = 0.0f;
    for (int s = wave; s <= NEGK; s += WAVES_PER_BLOCK) {
        // Stage 16 X rows, fully unrolled -> up to 16 global_load_b128 in flight.
        #pragma unroll
        for (int m = 0; m < 16; ++m) {
            const int idx = (s == 0) ? target[b0 + m]
                                     : noise[(size_t)(b0 + m) * NEGK + (s - 1)];
            const float4 v = ((const float4*)(emb_w + (size_t)idx * DIMD))[lane];
            float* dst = xb + m * PAD + lane * 4;
            dst[0] = v.x; dst[1] = v.y; dst[2] = v.z; dst[3] = v.w;
        }
        __threadfence_block();   // order xb stores before wave's LDS reads

        // S = X * C^T via 32 chained V_WMMA_F32_16X16X4_F32.
        // A lane(h,m) holds x_m[4kc+2h .. +1]; B lane(h,m) holds c_m[4kc+2h .. +1].
        v8f acc = {0.f, 0.f, 0.f, 0.f, 0.f, 0.f, 0.f, 0.f};
        const float* xrow = xb   + mm * PAD + 2 * h;
        const float* crow = cbuf + mm * PAD + 2 * h;
        #pragma unroll
        for (int kc = 0; kc < DIMD / 4; ++kc) {
            v2f a = *(const v2f*)(xrow + kc * 4);
            v2f b = *(const v2f*)(crow + kc * 4);
            acc = __builtin_amdgcn_wmma_f32_16x16x4_f32(
                false, a, false, b, (short)0, acc, false, false);
        }

        // Diagonal: row i -> VGPR i, lane i (i<8); VGPR i-8, lane i+16 (i>=8).
        const int  j      = (lane < 8) ? lane : (lane - 24);
        const bool active = (lane < 8) || (lane >= 24);
        float dv = acc[0];
        dv = (j == 1) ? acc[1] : dv;
        dv = (j == 2) ? acc[2] : dv;
        dv = (j == 3) ? acc[3] : dv;
        dv = (j == 4) ? acc[4] : dv;
        dv = (j == 5) ? acc[5] : dv;
        dv = (j == 6) ? acc[6] : dv;
        dv = (j == 7) ? acc[7] : dv;
        if (active) lsum += log_sigmoid(dv);
        __threadfence_block();   // order wave's LDS reads before next restage
    }

    // Wave reduce, then deterministic fixed-order block combine.
    #pragma unroll
    for (int off = 16; off > 0; off >>= 1)
        lsum += __shfl_xor(lsum, off, 32);
    if (lane == 0) wsum[wave] = lsum;
    __syncthreads();
    if (threadIdx.x == 0) {
        float t = 0.0f;
        #pragma unroll
        for (int w = 0; w < WAVES_PER_BLOCK; ++w) t += wsum[w];
        partial[tile] = t;
    }
}

__global__ void __launch_bounds__(1024)
cbow_reduce(const float* __restrict__ partial, int n, float invB,
            float* __restrict__ out) {
    __shared__ float sm[1024];
    const int t = threadIdx.x;
    float v = 0.0f;
    for (int i = t; i < n; i += 1024) v += partial[i];
    sm[t] = v;
    __syncthreads();
    for (int sft = 512; sft > 0; sft >>= 1) {
        if (t < sft) sm[t] += sm[t + sft];
        __syncthreads();
    }
    // loss = -(sum_b pos_ls + sum_{b,k} neg_ls) / B
    if (t == 0) out[0] = -sm[0] * invB;
}

extern "C" void kernel_launch(void* const* d_in, const int* in_sizes, int n_in,
                              void* d_out, int out_size, void* d_ws, size_t ws_size,
                              hipStream_t stream) {
    const int*   context = (const int*)d_in[0];
    const int*   target  = (const int*)d_in[1];
    const int*   noise   = (const int*)d_in[2];
    const float* emb_w   = (const float*)d_in[3];
    const float* ctx_w   = (const float*)d_in[4];
    float*       out     = (float*)d_out;

    const int B     = in_sizes[1];          // target has B elements
    const int tiles = B / 16;               // B = 16384 -> 1024 tiles/blocks
    float* partial  = (float*)d_ws;         // tiles * 4 bytes of scratch

    cbow_main<<<tiles, TPB, 0, stream>>>(context, target, noise, emb_w, ctx_w,
                                         partial, B);
    cbow_reduce<<<1, 1024, 0, stream>>>(partial, tiles, 1.0f / (float)B, out);
}